// PLATE_9371618640056
// MI455X (gfx1250) — compile-verified
//
#include <hip/hip_runtime.h>
#include <hip/hip_bf16.h>
#include <math.h>

typedef __attribute__((ext_vector_type(2))) float v2f;
typedef __attribute__((ext_vector_type(8))) float v8f;

#define NTHREADS 256

struct Params {
  const int *user_id, *target_item_id, *is_cold;
  const float *user_features, *item_features;
  const float *item_table, *user_table, *prompt_table;
  const float *item_fm1[6], *user_fm1[6];
  const float *item_fm2[6], *user_fm2[6];
  const float *fm1_w, *fm1_b;
  const float *dense_w, *dense_b;
  const float *ad1_w, *ad1_b, *adbn1_g, *adbn1_be, *adbn1_m, *adbn1_v;
  const float *ad2_w, *ad2_b, *adbn2_g, *adbn2_be, *adbn2_m, *adbn2_v;
  const float *ad3_w, *ad3_b, *ad_meta;
  const float *l1_w, *l1_b, *bn1_g, *bn1_be, *bn1_m, *bn1_v;
  const float *l2_w, *l2_b, *bn2_g, *bn2_be, *bn2_m, *bn2_v;
  const float *l3_w;
  float *out;
};

__device__ __forceinline__ float bn_apply(float x, const float* g, const float* be,
                                          const float* m, const float* v, int i) {
  return g[i] * (x - m[i]) / sqrtf(v[i] + 1e-5f) + be[i];
}

__global__ __launch_bounds__(NTHREADS) void deepfm_fused_kernel(Params p) {
  __shared__ float s_dnnx[16][1025];   // [row][0:64 domain | 64:128 prompt | 128:1024 stack]
  __shared__ float s_ifeat[16][28];
  __shared__ float s_ufeat[16][28];
  __shared__ float s_fm2acc[16];
  __shared__ float s_fm1[16];
  __shared__ float s_h1[16][32];
  __shared__ float s_h2[16][32];
  __shared__ float s_logit[16][10];
  __shared__ float s_prob[16][10];
  __shared__ float s_cpart[8][16][16];
  __shared__ float s_h1b[16][33];
  __shared__ float s_h2b[16][33];

  const int t = threadIdx.x;
  const int row0 = blockIdx.x * 16;

  // ---------------- Phase 0: stage features, init reducers ----------------
  for (int u = t; u < 16 * 28; u += NTHREADS) {
    int r = u / 28, c = u % 28;
    s_ifeat[r][c] = p.item_features[(row0 + r) * 28 + c];
    s_ufeat[r][c] = p.user_features[(row0 + r) * 28 + c];
  }
  if (t < 16) s_fm2acc[t] = 0.f;
  __syncthreads();

  // ---------------- Phase 1: 14 stack fields + FM2 partial ----------------
  {
    const int d  = t & 63;
    const int rg = t >> 6;
    for (int rr = 0; rr < 4; ++rr) {
      const int row  = rg + rr * 4;
      const int grow = row0 + row;
      float vals[14];
      #pragma unroll
      for (int j = 0; j < 4; ++j) {
        int id = (int)s_ifeat[row][4 + j];
        vals[j] = p.item_fm2[j][id * 64 + d];
      }
      {
        float v0 = 0.f, v1 = 0.f;
        #pragma unroll
        for (int q = 0; q < 10; ++q) {
          v0 += s_ifeat[row][8 + q]  * p.item_fm2[4][(1 + q) * 64 + d];
          v1 += s_ifeat[row][18 + q] * p.item_fm2[5][(1 + q) * 64 + d];
        }
        vals[4] = v0; vals[5] = v1;
      }
      #pragma unroll
      for (int j = 0; j < 4; ++j) {
        int id = (int)s_ufeat[row][4 + j];
        vals[6 + j] = p.user_fm2[j][id * 64 + d];
      }
      {
        float v0 = 0.f, v1 = 0.f;
        #pragma unroll
        for (int q = 0; q < 10; ++q) {
          v0 += s_ufeat[row][8 + q]  * p.user_fm2[4][(1 + q) * 64 + d];
          v1 += s_ufeat[row][18 + q] * p.user_fm2[5][(1 + q) * 64 + d];
        }
        vals[10] = v0; vals[11] = v1;
      }
      vals[12] = p.item_table[(long)p.target_item_id[grow] * 64 + d];
      vals[13] = p.user_table[(long)p.user_id[grow] * 64 + d];

      float se = 0.f, ssq = 0.f;
      #pragma unroll
      for (int f = 0; f < 14; ++f) {
        float x = vals[f];
        s_dnnx[row][128 + f * 64 + d] = x;
        se += x; ssq += x * x;
      }
      atomicAdd(&s_fm2acc[row], 0.5f * (se * se - ssq));
    }
  }

  // ---------------- Phase 2: FM first order ----------------
  if (t < 16) {
    const int row = t;
    float acc = p.fm1_b[0];
    #pragma unroll
    for (int j = 0; j < 4; ++j) {
      acc += p.item_fm1[j][(int)s_ifeat[row][4 + j]];
      acc += p.user_fm1[j][(int)s_ufeat[row][4 + j]];
    }
    #pragma unroll
    for (int q = 0; q < 10; ++q) {
      acc += s_ifeat[row][8 + q]  * p.item_fm1[4][1 + q];
      acc += s_ifeat[row][18 + q] * p.item_fm1[5][1 + q];
      acc += s_ufeat[row][8 + q]  * p.user_fm1[4][1 + q];
      acc += s_ufeat[row][18 + q] * p.user_fm1[5][1 + q];
    }
    #pragma unroll
    for (int k = 0; k < 4; ++k) {
      acc += s_ifeat[row][k] * p.fm1_w[k];
      acc += s_ufeat[row][k] * p.fm1_w[4 + k];
    }
    s_fm1[row] = acc;
  }
  __syncthreads();

  // ---------------- Phase 3: autodis MLP + softmax + prompt/domain ----------------
  for (int u = t; u < 512; u += NTHREADS) {         // layer1: user_emb(64) -> 32
    int row = u >> 5, o = u & 31;
    float a = p.ad1_b[o];
    for (int k = 0; k < 64; ++k) a += s_dnnx[row][960 + k] * p.ad1_w[k * 32 + o];
    a = bn_apply(a, p.adbn1_g, p.adbn1_be, p.adbn1_m, p.adbn1_v, o);
    s_h1[row][o] = fmaxf(a, 0.f);
  }
  __syncthreads();
  for (int u = t; u < 512; u += NTHREADS) {         // layer2: 32 -> 32
    int row = u >> 5, o = u & 31;
    float a = p.ad2_b[o];
    for (int k = 0; k < 32; ++k) a += s_h1[row][k] * p.ad2_w[k * 32 + o];
    a = bn_apply(a, p.adbn2_g, p.adbn2_be, p.adbn2_m, p.adbn2_v, o);
    s_h2[row][o] = fmaxf(a, 0.f);
  }
  __syncthreads();
  if (t < 160) {                                    // layer3: 32 -> 10 (pre-scaled by 1/TEMP)
    int row = t / 10, o = t % 10;
    float a = p.ad3_b[o];
    for (int k = 0; k < 32; ++k) a += s_h2[row][k] * p.ad3_w[k * 10 + o];
    s_logit[row][o] = a * 100.0f;
  }
  __syncthreads();
  if (t < 16) {                                     // softmax over 10
    float m = s_logit[t][0];
    for (int o = 1; o < 10; ++o) m = fmaxf(m, s_logit[t][o]);
    float e[10]; float s = 0.f;
    for (int o = 0; o < 10; ++o) { e[o] = expf(s_logit[t][o] - m); s += e[o]; }
    float inv = 1.f / s;
    for (int o = 0; o < 10; ++o) s_prob[t][o] = e[o] * inv;
  }
  __syncthreads();
  {
    const int d = t & 63, rg = t >> 6;
    for (int rr = 0; rr < 4; ++rr) {
      int row = rg + rr * 4, grow = row0 + row;
      float up = 0.f;
      #pragma unroll
      for (int o = 0; o < 10; ++o) up += s_prob[row][o] * p.ad_meta[o * 64 + d];
      s_dnnx[row][64 + d] = up;                                   // user_prompt
      s_dnnx[row][d] = p.prompt_table[p.is_cold[grow] * 64 + d];  // domain
    }
  }
  __syncthreads();

  // ---------------- Phase 4: dnn_x = dnn_in + relu(dense @ Wd + bd) ----------------
  for (int u = t; u < 16 * 1024; u += NTHREADS) {
    int row = u >> 10, c = u & 1023;
    float a = p.dense_b[c];
    #pragma unroll
    for (int k = 0; k < 4; ++k) a += s_ifeat[row][k] * p.dense_w[k * 1024 + c];
    #pragma unroll
    for (int k = 0; k < 4; ++k) a += s_ufeat[row][k] * p.dense_w[(4 + k) * 1024 + c];
    s_dnnx[row][c] += fmaxf(a, 0.f);
  }
  __syncthreads();

  // ---------------- Phase 5: WMMA  C[16,32] = dnn_x[16,1024] @ W1[1024,32] ----------------
  {
    const int wv = t >> 5, lane = t & 31;
    const int nt = wv & 1, ks = wv >> 1;           // 2 N-tiles x 4 K-slices
    const int lh = lane >> 4, lm = lane & 15;
    const int n  = nt * 16 + lm;
    const int k0 = ks * 256;
    v8f acc = {};
    for (int k = k0; k < k0 + 256; k += 4) {
      const int ka = k + 2 * lh;
      v2f a, b;
      a.x = s_dnnx[lm][ka];
      a.y = s_dnnx[lm][ka + 1];
      b.x = p.l1_w[ka * 32 + n];
      b.y = p.l1_w[(ka + 1) * 32 + n];
      acc = __builtin_amdgcn_wmma_f32_16x16x4_f32(false, a, false, b,
                                                  (short)0, acc, false, false);
    }
    #pragma unroll
    for (int r = 0; r < 8; ++r) s_cpart[wv][r + 8 * lh][lm] = acc[r];
  }
  __syncthreads();
  for (int u = t; u < 512; u += NTHREADS) {        // reduce K-slices + bias + BN + ReLU
    int row = u >> 5, o = u & 31;
    int nt2 = o >> 4, nn = o & 15;
    float a = s_cpart[nt2][row][nn] + s_cpart[2 + nt2][row][nn]
            + s_cpart[4 + nt2][row][nn] + s_cpart[6 + nt2][row][nn];
    a += p.l1_b[o];
    a = bn_apply(a, p.bn1_g, p.bn1_be, p.bn1_m, p.bn1_v, o);
    s_h1b[row][o] = fmaxf(a, 0.f);
  }
  __syncthreads();

  // ---------------- Phase 6: WMMA  C[16,32] = h1[16,32] @ W2[32,32] ----------------
  {
    const int wv = t >> 5, lane = t & 31;
    if (wv < 2) {
      const int lh = lane >> 4, lm = lane & 15;
      const int n = wv * 16 + lm;
      v8f acc = {};
      for (int k = 0; k < 32; k += 4) {
        const int ka = k + 2 * lh;
        v2f a, b;
        a.x = s_h1b[lm][ka];
        a.y = s_h1b[lm][ka + 1];
        b.x = p.l2_w[ka * 32 + n];
        b.y = p.l2_w[(ka + 1) * 32 + n];
        acc = __builtin_amdgcn_wmma_f32_16x16x4_f32(false, a, false, b,
                                                    (short)0, acc, false, false);
      }
      #pragma unroll
      for (int r = 0; r < 8; ++r) s_cpart[wv][r + 8 * lh][lm] = acc[r];
    }
  }
  __syncthreads();
  for (int u = t; u < 512; u += NTHREADS) {        // bias + BN + ReLU
    int row = u >> 5, o = u & 31;
    float a = s_cpart[o >> 4][row][o & 15] + p.l2_b[o];
    a = bn_apply(a, p.bn2_g, p.bn2_be, p.bn2_m, p.bn2_v, o);
    s_h2b[row][o] = fmaxf(a, 0.f);
  }
  __syncthreads();

  // ---------------- Phase 7: final dot + sigmoid ----------------
  if (t < 16) {
    float a = 0.f;
    for (int o = 0; o < 32; ++o) a += s_h2b[t][o] * p.l3_w[o];
    float z = s_fm1[t] + s_fm2acc[t] + a;
    p.out[row0 + t] = 1.f / (1.f + expf(-z));
  }
}

extern "C" void kernel_launch(void* const* d_in, const int* in_sizes, int n_in,
                              void* d_out, int out_size, void* d_ws, size_t ws_size,
                              hipStream_t stream) {
  (void)n_in; (void)out_size; (void)d_ws; (void)ws_size;
  Params P;
  P.user_id        = (const int*)d_in[0];
  P.target_item_id = (const int*)d_in[1];
  // d_in[2] history_item_id, d_in[3] history_len: unused by the reference forward
  P.user_features  = (const float*)d_in[4];
  P.item_features  = (const float*)d_in[5];
  P.is_cold        = (const int*)d_in[6];
  P.item_table     = (const float*)d_in[7];
  P.user_table     = (const float*)d_in[8];
  P.prompt_table   = (const float*)d_in[9];
  for (int j = 0; j < 6; ++j) {
    P.item_fm1[j] = (const float*)d_in[10 + j];
    P.user_fm1[j] = (const float*)d_in[16 + j];
    P.item_fm2[j] = (const float*)d_in[22 + j];
    P.user_fm2[j] = (const float*)d_in[28 + j];
  }
  P.fm1_w   = (const float*)d_in[34];
  P.fm1_b   = (const float*)d_in[35];
  P.dense_w = (const float*)d_in[36];
  P.dense_b = (const float*)d_in[37];
  P.ad1_w   = (const float*)d_in[38];
  P.ad1_b   = (const float*)d_in[39];
  P.adbn1_g = (const float*)d_in[40];
  P.adbn1_be= (const float*)d_in[41];
  P.adbn1_m = (const float*)d_in[42];
  P.adbn1_v = (const float*)d_in[43];
  P.ad2_w   = (const float*)d_in[44];
  P.ad2_b   = (const float*)d_in[45];
  P.adbn2_g = (const float*)d_in[46];
  P.adbn2_be= (const float*)d_in[47];
  P.adbn2_m = (const float*)d_in[48];
  P.adbn2_v = (const float*)d_in[49];
  P.ad3_w   = (const float*)d_in[50];
  P.ad3_b   = (const float*)d_in[51];
  P.ad_meta = (const float*)d_in[52];
  P.l1_w    = (const float*)d_in[53];
  P.l1_b    = (const float*)d_in[54];
  P.bn1_g   = (const float*)d_in[55];
  P.bn1_be  = (const float*)d_in[56];
  P.bn1_m   = (const float*)d_in[57];
  P.bn1_v   = (const float*)d_in[58];
  P.l2_w    = (const float*)d_in[59];
  P.l2_b    = (const float*)d_in[60];
  P.bn2_g   = (const float*)d_in[61];
  P.bn2_be  = (const float*)d_in[62];
  P.bn2_m   = (const float*)d_in[63];
  P.bn2_v   = (const float*)d_in[64];
  P.l3_w    = (const float*)d_in[65];
  P.out     = (float*)d_out;

  const int B = in_sizes[0];          // 16384
  const int nblocks = B / 16;         // 16 rows per workgroup
  deepfm_fused_kernel<<<nblocks, NTHREADS, 0, stream>>>(P);
}